// CosineLoss_52166672777434
// MI455X (gfx1250) — compile-verified
//
#include <hip/hip_runtime.h>

typedef __attribute__((ext_vector_type(2))) float v2f;
typedef __attribute__((ext_vector_type(4))) float v4f;
typedef __attribute__((ext_vector_type(8))) float v8f;

#define D_DIM 512

// One wave32 per 16-row tile. V_WMMA_F32_16X16X4_F32 forms the three 16x16
// Gram matrices (ehr*cxr^T, ehr*ehr^T, cxr*cxr^T); diagonals give the per-row
// dot products. Each b128 load feeds two WMMA K-steps:
//   lanes 0-15 hold cols {8s..8s+3}, lanes 16-31 hold cols {8s+4..8s+7};
//   step 1 uses .xy (K-group {8s,8s+1,8s+4,8s+5}),
//   step 2 uses .zw (K-group {8s+2,8s+3,8s+6,8s+7}).
// Any partition of columns into K-groups is valid because A and B share the
// same lane addressing, so products pair on matching columns.
__global__ __launch_bounds__(256) void CosineLoss_tile_kernel(
    const float* __restrict__ cxr, const float* __restrict__ ehr,
    float* __restrict__ partial, int nTiles) {
  const int lane = threadIdx.x & 31;
  const int waveInBlock = threadIdx.x >> 5;
  const int tile = blockIdx.x * (blockDim.x >> 5) + waveInBlock;

  __shared__ float wsum[8];

  float tile_sum = 0.0f;

  if (tile < nTiles) {
    const int row0 = tile * 16;
    const int r = lane & 15;          // row within tile (both lane halves)
    const int ch = (lane >> 4) << 2;  // col sub-offset: 0 (lanes 0-15) / 4

    const float* __restrict__ ap = ehr + (size_t)(row0 + r) * D_DIM + ch;
    const float* __restrict__ bp = cxr + (size_t)(row0 + r) * D_DIM + ch;

    v8f acc_ab = {};  // diag -> dot(ehr_i, cxr_i)
    v8f acc_aa = {};  // diag -> ||ehr_i||^2
    v8f acc_bb = {};  // diag -> ||cxr_i||^2

#pragma unroll 4
    for (int k = 0; k < D_DIM; k += 8) {
      v4f a = *(const v4f*)(ap + k);
      v4f b = *(const v4f*)(bp + k);
      v2f a0 = a.xy, a1 = a.zw;
      v2f b0 = b.xy, b1 = b.zw;
      acc_ab = __builtin_amdgcn_wmma_f32_16x16x4_f32(
          false, a0, false, b0, (short)0, acc_ab, false, false);
      acc_aa = __builtin_amdgcn_wmma_f32_16x16x4_f32(
          false, a0, false, a0, (short)0, acc_aa, false, false);
      acc_bb = __builtin_amdgcn_wmma_f32_16x16x4_f32(
          false, b0, false, b0, (short)0, acc_bb, false, false);
      acc_ab = __builtin_amdgcn_wmma_f32_16x16x4_f32(
          false, a1, false, b1, (short)0, acc_ab, false, false);
      acc_aa = __builtin_amdgcn_wmma_f32_16x16x4_f32(
          false, a1, false, a1, (short)0, acc_aa, false, false);
      acc_bb = __builtin_amdgcn_wmma_f32_16x16x4_f32(
          false, b1, false, b1, (short)0, acc_bb, false, false);
    }

    // Diagonal (i,i) of a 16x16 f32 C/D matrix lives in acc VGPR (i & 7) at
    // lane i (i < 8) or lane (i - 8) + 24 (i >= 8). Sum the 16 cosines.
#pragma unroll
    for (int g = 0; g < 8; ++g) {
      float ab = acc_ab[g];
      float aa = acc_aa[g];
      float bb = acc_bb[g];
      float c = ab * rsqrtf(aa * bb);
      tile_sum += __shfl(c, g, 32) + __shfl(c, 24 + g, 32);
    }
  }

  if (lane == 0) wsum[waveInBlock] = tile_sum;
  __syncthreads();

  if (threadIdx.x == 0) {
    float s = 0.0f;
#pragma unroll
    for (int i = 0; i < 8; ++i) s += wsum[i];
    partial[blockIdx.x] = s;
  }
}

__global__ __launch_bounds__(256) void CosineLoss_finalize_kernel(
    const float* __restrict__ partial, int nPartial, float* __restrict__ out,
    float invN) {
  __shared__ float sdata[256];
  float s = 0.0f;
  for (int i = threadIdx.x; i < nPartial; i += 256) s += partial[i];
  sdata[threadIdx.x] = s;
  __syncthreads();
#pragma unroll
  for (int off = 128; off > 0; off >>= 1) {
    if ((int)threadIdx.x < off) sdata[threadIdx.x] += sdata[threadIdx.x + off];
    __syncthreads();
  }
  if (threadIdx.x == 0) out[0] = 1.0f - sdata[0] * invN;
}

extern "C" void kernel_launch(void* const* d_in, const int* in_sizes, int n_in,
                              void* d_out, int out_size, void* d_ws,
                              size_t ws_size, hipStream_t stream) {
  (void)n_in; (void)out_size; (void)ws_size;
  const float* cxr = (const float*)d_in[0];
  const float* ehr = (const float*)d_in[1];
  float* out = (float*)d_out;

  const int N = in_sizes[0] / D_DIM;       // 65536
  const int nTiles = N / 16;               // 16 rows per wave
  const int wavesPerBlock = 8;             // 256 threads, wave32
  const int nBlocks = (nTiles + wavesPerBlock - 1) / wavesPerBlock;

  float* partial = (float*)d_ws;           // nBlocks floats of scratch

  CosineLoss_tile_kernel<<<nBlocks, 256, 0, stream>>>(cxr, ehr, partial,
                                                      nTiles);
  CosineLoss_finalize_kernel<<<1, 256, 0, stream>>>(partial, nBlocks, out,
                                                    1.0f / (float)N);
}